// DetectionLoss_68779606279019
// MI455X (gfx1250) — compile-verified
//
#include <hip/hip_runtime.h>
#include <math.h>

// ---------------- problem constants (from reference) ----------------
#define B_    32
#define N_    16384
#define M_    128
#define C_    86          // 5 + NUM_CLASSES columns per prediction
#define NL_   80          // NUM_LOGITS = 5 + 81 - 6
#define NEG_  (-1.0e9f)
#define IOU_THR   0.4f
#define AREA_THR  0.01f
#define SMALL_W   1.5f

#define TPB   256                       // 8 wave32s per block
#define BLKS_PER_IMG (N_ / TPB)         // 64 blocks per image
#define NWAVES (TPB / 32)

// ---------------- CDNA5 async global->LDS path ----------------
#if defined(__HIP_DEVICE_COMPILE__) && \
    __has_builtin(__builtin_amdgcn_global_load_async_to_lds_b128) && \
    __has_builtin(__builtin_amdgcn_s_wait_asynccnt)
#define USE_ASYNC_LDS 1
#else
#define USE_ASYNC_LDS 0
#endif

// ---------------- CDNA5 Tensor Data Mover path ----------------
#if defined(__HIP_DEVICE_COMPILE__) && \
    __has_builtin(__builtin_amdgcn_tensor_load_to_lds) && \
    __has_builtin(__builtin_amdgcn_s_wait_tensorcnt)
#define USE_TDM 1
#else
#define USE_TDM 0
#endif

typedef int v4i __attribute__((ext_vector_type(4)));
typedef __attribute__((address_space(1))) v4i* as1_v4i_p;
typedef __attribute__((address_space(3))) v4i* as3_v4i_p;

#if USE_TDM
typedef unsigned uint32x4 __attribute__((ext_vector_type(4)));
typedef int      int32x8  __attribute__((ext_vector_type(8)));
typedef int      int32x4  __attribute__((ext_vector_type(4)));

// 2D TDM load: tile_d1 rows of tile_d0 4-byte elements, row stride = stride0
// elements, packed contiguously into LDS (ISA 8.2 load pseudocode).
__device__ __forceinline__ void tdm_load_2d(const void* gptr, void* lds,
                                            unsigned tensor_d0, unsigned tensor_d1,
                                            unsigned tile_d0, unsigned tile_d1,
                                            unsigned long long stride0_elems) {
  const unsigned long long ga = (unsigned long long)(uintptr_t)gptr;   // 57-bit VA
  const unsigned ldsOff = (unsigned)(uintptr_t)lds;  // low 32 bits of generic = LDS offset
  uint32x4 g0;
  g0.x = 0x1u;                                        // count=1, user mode, no gather
  g0.y = ldsOff;                                      // D#.lds_addr
  g0.z = (unsigned)(ga & 0xFFFFFFFFull);              // global_addr[31:0]
  g0.w = ((unsigned)(ga >> 32) & 0x01FFFFFFu)         // global_addr[56:32]
         | (2u << 30);                                // type=2 ("image")
  int32x8 g1;
  g1[0] = (int)(2u << 16);                            // wg_mask=0, data_size=2 (4B)
  g1[1] = (int)((tensor_d0 & 0xFFFFu) << 16);         // abar_addr=0 | tensor_dim0[15:0]
  g1[2] = (int)(((tensor_d0 >> 16) & 0xFFFFu) | ((tensor_d1 & 0xFFFFu) << 16));
  g1[3] = (int)(((tensor_d1 >> 16) & 0xFFFFu) | ((tile_d0 & 0xFFFFu) << 16));
  g1[4] = (int)(tile_d1 & 0xFFFFu);                   // tile_dim1 | tile_dim2=0
  g1[5] = (int)(unsigned)(stride0_elems & 0xFFFFFFFFull);    // dim0_stride[31:0]
  g1[6] = (int)(unsigned)((stride0_elems >> 32) & 0xFFFFull);// dim0_stride[47:32] | d1s=0
  g1[7] = 0;                                          // tensor_dim1_stride hi (unused, 2D)
  const int32x4 z4 = {0, 0, 0, 0};                    // groups 2/3: zeros (<=2D tensor)
  const int32x8 z8 = {0, 0, 0, 0, 0, 0, 0, 0};        // aux operand (6-arg toolchain)
  __builtin_amdgcn_tensor_load_to_lds(g0, g1, z4, z4, z8, 0);
}
#endif

// Order-preserving float -> uint key: atomicMax on key == float max, and
// key equality == exact float equality.
__device__ __forceinline__ unsigned fkey(float f) {
  unsigned u = __float_as_uint(f);
  return (u & 0x80000000u) ? ~u : (u | 0x80000000u);
}

// Stage per-block data into LDS:
//  sT[640]  : 128x5 target tile (async b128 loads, ASYNCcnt)
//  sP[1280] : 256x5 box+conf tile gathered from 86-wide rows (TDM, TENSORcnt)
__device__ __forceinline__ void stage_tiles(const float* __restrict__ tg,
                                            const float* __restrict__ predRow0,
                                            float* sT, float* sP) {
  const int t = threadIdx.x;
#if USE_TDM
  if (t < 32) {   // wave 0 issues the DMA once; EXEC is ignored by TDM
    tdm_load_2d(predRow0, sP, /*tensor_d0=*/C_, /*tensor_d1=*/TPB,
                /*tile_d0=*/5, /*tile_d1=*/TPB, /*stride0=*/C_);
  }
#else
  {
    const float* r = predRow0 + (size_t)t * C_;
    float2 a = *(const float2*)r, b2 = *(const float2*)(r + 2);
    float* dst = sP + t * 5;
    dst[0] = a.x; dst[1] = a.y; dst[2] = b2.x; dst[3] = b2.y; dst[4] = r[4];
  }
#endif
#if USE_ASYNC_LDS
  if (t < (M_ * 5 * 4) / 16) {   // 160 x 16B chunks
    __builtin_amdgcn_global_load_async_to_lds_b128(
        (as1_v4i_p)(tg + 4 * t), (as3_v4i_p)(sT + 4 * t), 0, 0);
  }
  __builtin_amdgcn_s_wait_asynccnt(0);   // each wave drains its own ASYNCcnt
#else
  for (int k = t; k < M_ * 5; k += TPB) sT[k] = tg[k];
#endif
#if USE_TDM
  if (t < 32) __builtin_amdgcn_s_wait_tensorcnt(0);
#endif
  __syncthreads();
}

// Best IoU + first-index argmax over the 128 targets (invalid -> NEG).
__device__ __forceinline__ void best_match(const float pb[4], const float* sT,
                                           float& bestIou, int& bestIdx) {
  float best = -INFINITY; int bi = 0;
  const float a1 = (pb[2] - pb[0]) * (pb[3] - pb[1]);
#pragma unroll 4
  for (int j = 0; j < M_; ++j) {
    const float g0 = sT[j*5+0], g1 = sT[j*5+1], g2 = sT[j*5+2], g3 = sT[j*5+3], gc = sT[j*5+4];
    float x1 = fmaxf(pb[0], g0);
    float y1 = fmaxf(pb[1], g1);
    float x2 = fminf(pb[2], g2);
    float y2 = fminf(pb[3], g3);
    float inter = fmaxf(x2 - x1, 0.0f) * fmaxf(y2 - y1, 0.0f);
    float a2 = (g2 - g0) * (g3 - g1);
    float iou = inter / (a1 + a2 - inter + 1e-6f);
    iou = (gc != -1.0f) ? iou : NEG_;
    if (iou > best) { best = iou; bi = j; }   // strict '>' keeps first max index
  }
  bestIou = best; bestIdx = bi;
}

// ---------------- kernel 0: zero ws accumulators ----------------
__global__ void k_init(unsigned* ws) {
  // layout: [0..31] matchedCnt, [32..63] maxKey, [64..255] acc (32 imgs x 6 f32)
  if (threadIdx.x < 256) ws[threadIdx.x] = 0u;
}

// ---------------- kernel 1: per-image match stats ----------------
__global__ void __launch_bounds__(TPB)
k_stats(const float* __restrict__ preds, const float* __restrict__ targets,
        unsigned* __restrict__ matchedCnt, unsigned* __restrict__ maxKey) {
  __shared__ float sT[M_ * 5];
  __shared__ float sP[TPB * 5];
  __shared__ unsigned sCnt[NWAVES];
  __shared__ unsigned sKey[NWAVES];

  const int b  = blockIdx.x / BLKS_PER_IMG;
  const int i0 = (blockIdx.x % BLKS_PER_IMG) * TPB;
  stage_tiles(targets + (size_t)b * M_ * 5,
              preds + ((size_t)b * N_ + i0) * C_, sT, sP);

  const float* p = sP + threadIdx.x * 5;
  const float pb[4] = {p[0], p[1], p[2], p[3]};
  float best; int bi;
  best_match(pb, sT, best, bi);

  unsigned cnt = (best > IOU_THR) ? 1u : 0u;
  unsigned key = fkey(best);
  for (int o = 16; o > 0; o >>= 1) {               // wave32 reduction
    cnt += __shfl_down(cnt, o);
    unsigned ok = __shfl_down(key, o);
    key = key > ok ? key : ok;
  }
  const int wid = threadIdx.x >> 5, lid = threadIdx.x & 31;
  if (lid == 0) { sCnt[wid] = cnt; sKey[wid] = key; }
  __syncthreads();
  if (threadIdx.x == 0) {
    unsigned c = 0, k = 0;
    for (int w = 0; w < NWAVES; ++w) { c += sCnt[w]; k = k > sKey[w] ? k : sKey[w]; }
    if (c) atomicAdd(&matchedCnt[b], c);
    atomicMax(&maxKey[b], k);
  }
}

// ---------------- kernel 2: masked loss accumulation ----------------
__global__ void __launch_bounds__(TPB)
k_accum(const float* __restrict__ preds, const float* __restrict__ targets,
        const unsigned* __restrict__ matchedCnt, const unsigned* __restrict__ maxKey,
        float* __restrict__ acc) {
  __shared__ float sT[M_ * 5];
  __shared__ float sP[TPB * 5];
  __shared__ float sRed[6][NWAVES];

  const int b  = blockIdx.x / BLKS_PER_IMG;
  const int i0 = (blockIdx.x % BLKS_PER_IMG) * TPB;
  const int i  = i0 + threadIdx.x;
  stage_tiles(targets + (size_t)b * M_ * 5,
              preds + ((size_t)b * N_ + i0) * C_, sT, sP);

  const float* p = sP + threadIdx.x * 5;
  const float pb[4] = {p[0], p[1], p[2], p[3]};
  const float conf  = p[4];
  float best; int bi;
  best_match(pb, sT, best, bi);

  const bool anyM = matchedCnt[b] > 0u;
  const unsigned mk = maxKey[b];
  const float mf = anyM ? ((best > IOU_THR) ? 1.0f : 0.0f)
                        : ((fkey(best) == mk) ? 1.0f : 0.0f);

  const float bce1 = -fmaxf(logf(conf), -100.0f);      // bce(conf, 1)
  const float bce0 = -fmaxf(log1pf(-conf), -100.0f);   // bce(conf, 0)

  float v[6];
  v[0] = mf;                    // n_m
  v[1] = 0.0f;                  // bbox sum
  v[2] = 0.0f;                  // nll sum
  v[3] = bce1 * mf;             // matched conf
  v[4] = bce0 * (1.0f - mf);    // unmatched conf
  v[5] = bce0;                  // empty-branch sum

  if (mf > 0.0f) {
    const float g0 = sT[bi*5+0], g1 = sT[bi*5+1], g2 = sT[bi*5+2], g3 = sT[bi*5+3], gc = sT[bi*5+4];
    const float area = (g2 - g0) * (g3 - g1);
    const float w = (area < AREA_THR) ? SMALL_W : 1.0f;
    const float gt[4] = {g0, g1, g2, g3};
    float s = 0.0f;
#pragma unroll
    for (int k = 0; k < 4; ++k) {
      float d = pb[k] - gt[k];
      float ad = fabsf(d);
      s += (ad < 1.0f) ? 0.5f * d * d : ad - 0.5f;
    }
    v[1] = s * w;

    int ci = (int)gc; ci = ci < 0 ? 0 : (ci > NL_ - 1 ? NL_ - 1 : ci);
    const float* lg = preds + ((size_t)b * N_ + i) * C_ + 6;  // 80 logits, 8B-aligned
    float mx = -INFINITY;
#pragma unroll 8
    for (int k = 0; k < NL_ / 2; ++k) {
      float2 q = *(const float2*)(lg + 2 * k);
      mx = fmaxf(mx, fmaxf(q.x, q.y));
    }
    float se = 0.0f;
#pragma unroll 8
    for (int k = 0; k < NL_ / 2; ++k) {
      float2 q = *(const float2*)(lg + 2 * k);
      se += expf(q.x - mx) + expf(q.y - mx);
    }
    v[2] = mx + logf(se) - lg[ci];             // -log_softmax[ci]
  }

  // block reduction: wave32 shfl, cross-wave via LDS, one atomic per block
  const int wid = threadIdx.x >> 5, lid = threadIdx.x & 31;
#pragma unroll
  for (int k = 0; k < 6; ++k) {
    float x = v[k];
    for (int o = 16; o > 0; o >>= 1) x += __shfl_down(x, o);
    if (lid == 0) sRed[k][wid] = x;
  }
  __syncthreads();
  if (threadIdx.x == 0) {
#pragma unroll
    for (int k = 0; k < 6; ++k) {
      float s = 0.0f;
      for (int w = 0; w < NWAVES; ++w) s += sRed[k][w];
      atomicAdd(&acc[b * 6 + k], s);
    }
  }
}

// ---------------- kernel 3: finalize (one wave32) ----------------
__global__ void k_final(const float* __restrict__ targets,
                        const float* __restrict__ acc, float* __restrict__ out) {
  const int b = threadIdx.x;   // one lane per image
  float per = 0.0f;
  if (b < B_) {
    bool validAny = false;
    const float* tg = targets + (size_t)b * M_ * 5;
    for (int j = 0; j < M_; ++j) validAny = validAny || (tg[j * 5 + 4] != -1.0f);
    const float* a = acc + b * 6;
    const float n_m      = a[0];
    const float n_m_safe = fmaxf(n_m, 1.0f);
    const float bbox_loss = a[1] / (n_m_safe * 4.0f);
    const float cls_loss  = a[2] / n_m_safe;
    const float m_conf    = a[3] / n_m_safe;
    const float n_u       = (float)N_ - n_m;
    const float u_conf    = a[4] / fmaxf(n_u, 1.0f);
    const float conf_loss = (n_u > 0.0f) ? 0.5f * (m_conf + u_conf) : m_conf;
    const float full  = bbox_loss * 5.0f + cls_loss + conf_loss * 2.0f;
    const float empty = a[5] / (float)N_;
    per = validAny ? full : empty;
  }
  for (int o = 16; o > 0; o >>= 1) per += __shfl_down(per, o);
  if (threadIdx.x == 0) out[0] = per / (float)B_;
}

// ---------------- host side ----------------
extern "C" void kernel_launch(void* const* d_in, const int* in_sizes, int n_in,
                              void* d_out, int out_size, void* d_ws, size_t ws_size,
                              hipStream_t stream) {
  const float* preds   = (const float*)d_in[0];   // (32, 16384, 86) f32
  const float* targets = (const float*)d_in[1];   // (32, 128, 5)    f32
  float* out = (float*)d_out;

  unsigned* ws         = (unsigned*)d_ws;
  unsigned* matchedCnt = ws;                 // 32 u32
  unsigned* maxKey     = ws + 32;            // 32 u32
  float*    acc        = (float*)(ws + 64);  // 32 x 6 f32

  k_init<<<1, 256, 0, stream>>>(ws);
  dim3 grid(B_ * BLKS_PER_IMG);
  k_stats<<<grid, TPB, 0, stream>>>(preds, targets, matchedCnt, maxKey);
  k_accum<<<grid, TPB, 0, stream>>>(preds, targets, matchedCnt, maxKey, acc);
  k_final<<<1, 32, 0, stream>>>(targets, acc, out);
}